// DataDrivenANFIS_73272142070110
// MI455X (gfx1250) — compile-verified
//
#include <hip/hip_runtime.h>
#include <hip/hip_bf16.h>

// ANFIS fused forward for MI455X (gfx1250), wave32 + V_WMMA_F32_16X16X4_F32.
// One wave handles a 16-row batch tile end-to-end:
//   firing (WMMA cross + exp) -> LDS -> WMMA consequent GEMM -> fold -> out.

#define NB 16384
#define NR 512
#define NF 32
#define NO 8
#define FN_STRIDE 516   // 512 + 4 pad floats: kills LDS bank conflicts
#define X_STRIDE  34    // 32 + 2 pad floats

typedef __attribute__((ext_vector_type(2))) float v2f;
typedef __attribute__((ext_vector_type(8))) float v8f;

static __device__ __forceinline__ v8f wmma_f32(v2f a, v2f b, v8f c) {
    // D(16x16,f32) += A(16x4,f32) * B(4x16,f32)
    return __builtin_amdgcn_wmma_f32_16x16x4_f32(
        /*neg_a=*/false, a, /*neg_b=*/false, b,
        /*c_mod=*/(short)0, c, /*reuse_a=*/false, /*reuse_b=*/false);
}

__launch_bounds__(32)
__global__ void anfis_wmma_kernel(const float* __restrict__ X,
                                  const float* __restrict__ Cc,   // centers [R,F]
                                  const float* __restrict__ Wd,   // widths  [R]
                                  const float* __restrict__ Wc,   // consequent_w [R,F,O]
                                  const float* __restrict__ Cb,   // consequent_b [R,O]
                                  float* __restrict__ Out)        // [B,O]
{
    extern __shared__ float lds[];
    float* fnl = lds;                  // unnormalized firing  [16][FN_STRIDE]
    float* xl  = lds + 16 * FN_STRIDE; // row-major X tile     [16][X_STRIDE]

    const int lane = threadIdx.x & 31;
    const int n16  = lane & 15;   // N (or M) index inside 16-group
    const int hf   = lane >> 4;   // which half of the wave
    const int b0   = blockIdx.x * 16;

    const v8f vz = {0.f, 0.f, 0.f, 0.f, 0.f, 0.f, 0.f, 0.f};

    // ---- 1. X tile: A-layout regs + LDS stash + |x|^2 ------------------
    // A 16x4 f32 layout: lanes 0-15 hold (M=lane, K={0,1}); lanes 16-31 (M, K={2,3}).
    v2f a[8];
    float x2p = 0.f;
    #pragma unroll
    for (int kb = 0; kb < 8; ++kb) {
        const float* p = X + (size_t)(b0 + n16) * NF + kb * 4 + 2 * hf;
        a[kb] = *reinterpret_cast<const v2f*>(p);
        x2p += a[kb].x * a[kb].x + a[kb].y * a[kb].y;
        xl[n16 * X_STRIDE + kb * 4 + 2 * hf]     = a[kb].x;
        xl[n16 * X_STRIDE + kb * 4 + 2 * hf + 1] = a[kb].y;
    }
    float x2 = x2p + __shfl_xor(x2p, 16, 32);        // lane -> |x_{n16}|^2
    float x2m[8];
    #pragma unroll
    for (int v = 0; v < 8; ++v)                      // C/D-row layout: M = v + 8*hf
        x2m[v] = __shfl(x2, v + 8 * hf, 32);

    // ---- 2. firing strengths over all 512 rules (WMMA cross) ----------
    float srow[8];
    #pragma unroll
    for (int v = 0; v < 8; ++v) srow[v] = 0.f;

    for (int rt = 0; rt < NR / 16; ++rt) {
        float c2p = 0.f;
        v8f acc = vz;
        #pragma unroll
        for (int kb = 0; kb < 8; ++kb) {
            // B 4x16: lane holds (K={2*hf,2*hf+1}, N=n16): B[k][n] = centers[rule=n][k]
            const float* p = Cc + (size_t)(rt * 16 + n16) * NF + kb * 4 + 2 * hf;
            v2f bm = *reinterpret_cast<const v2f*>(p);
            c2p += bm.x * bm.x + bm.y * bm.y;
            acc = wmma_f32(a[kb], bm, acc);
        }
        float c2 = c2p + __shfl_xor(c2p, 16, 32);    // lane -> |c_{rt*16+n16}|^2
        float w  = Wd[rt * 16 + n16];
        float s  = fabsf(w) + 0.1f;
        // fast reciprocal (v_rcp_f32) instead of the IEEE div sequence
        float coef = -0.5f * __builtin_amdgcn_rcpf(s * s);
        #pragma unroll
        for (int v = 0; v < 8; ++v) {
            float dist = x2m[v] + c2 - 2.0f * acc[v];
            float fr   = __expf(dist * coef);        // unnormalized firing
            fnl[(v + 8 * hf) * FN_STRIDE + rt * 16 + n16] = fr;
            srow[v] += fr;                           // per-lane partial over N
        }
    }
    float rsv[8];
    #pragma unroll
    for (int v = 0; v < 8; ++v) {                    // reduce over 16-lane group
        float t = srow[v];
        t += __shfl_xor(t, 1, 32);
        t += __shfl_xor(t, 2, 32);
        t += __shfl_xor(t, 4, 32);
        t += __shfl_xor(t, 8, 32);
        rsv[v] = __builtin_amdgcn_rcpf(t + 1e-8f);   // fold normalization at the end
    }

    // ---- 3. consequent GEMM: FIRE[16,512] x W2[512,256], fused fold ---
    float outA[8];
    #pragma unroll
    for (int v = 0; v < 8; ++v) outA[v] = 0.f;

    for (int nt = 0; nt < 16; ++nt) {                // N-tiles over F*O = 256 cols
        v8f acc0 = vz, acc1 = vz;
        #pragma unroll 4
        for (int kb = 0; kb < 128; kb += 2) {        // K = 512 rules, 4 per WMMA
            int K0 = kb * 4 + 2 * hf;
            v2f a0 = *reinterpret_cast<const v2f*>(&fnl[n16 * FN_STRIDE + kb * 4 + 2 * hf]);
            v2f a1 = *reinterpret_cast<const v2f*>(&fnl[n16 * FN_STRIDE + (kb + 1) * 4 + 2 * hf]);
            v2f b0v, b1v;
            b0v.x = Wc[(size_t)(K0    ) * 256 + nt * 16 + n16];
            b0v.y = Wc[(size_t)(K0 + 1) * 256 + nt * 16 + n16];
            b1v.x = Wc[(size_t)(K0 + 4) * 256 + nt * 16 + n16];
            b1v.y = Wc[(size_t)(K0 + 5) * 256 + nt * 16 + n16];
            acc0 = wmma_f32(a0, b0v, acc0);
            acc1 = wmma_f32(a1, b1v, acc1);
        }
        // fold: out[b, o=n16%8] += x[b, f=2*nt + n16/8] * G
        #pragma unroll
        for (int v = 0; v < 8; ++v) {
            float xa = xl[(v + 8 * hf) * X_STRIDE + 2 * nt + (n16 >> 3)];
            outA[v] += (acc0[v] + acc1[v]) * xa;
        }
    }

    // ---- 4. bias GEMM tile: FIRE x consequent_b (cols 8..15 zero) -----
    {
        v8f acc = vz;
        #pragma unroll 4
        for (int kb = 0; kb < 128; ++kb) {
            int K = kb * 4 + 2 * hf;
            v2f av = *reinterpret_cast<const v2f*>(&fnl[n16 * FN_STRIDE + kb * 4 + 2 * hf]);
            v2f bv;
            bv.x = (n16 < NO) ? Cb[(size_t)K * NO + n16]       : 0.f;
            bv.y = (n16 < NO) ? Cb[(size_t)(K + 1) * NO + n16] : 0.f;
            acc = wmma_f32(av, bv, acc);
        }
        #pragma unroll
        for (int v = 0; v < 8; ++v)
            outA[v] += acc[v];   // lanes with n16>=8 carry exact zeros
    }

    // ---- 5. combine f-pairs, normalize, store -------------------------
    #pragma unroll
    for (int v = 0; v < 8; ++v) {
        float r = outA[v] + __shfl_xor(outA[v], 8, 32);  // sum the two f per o
        r *= rsv[v];
        if (n16 < NO)
            Out[(size_t)(b0 + v + 8 * hf) * NO + n16] = r;
    }
}

extern "C" void kernel_launch(void* const* d_in, const int* in_sizes, int n_in,
                              void* d_out, int out_size, void* d_ws, size_t ws_size,
                              hipStream_t stream) {
    const float* X  = (const float*)d_in[0];  // inputs       [B,F]
    const float* Cc = (const float*)d_in[1];  // centers      [R,F]
    const float* Wd = (const float*)d_in[2];  // widths       [R]
    const float* Wc = (const float*)d_in[3];  // consequent_w [R,F,O]
    const float* Cb = (const float*)d_in[4];  // consequent_b [R,O]
    float* Out = (float*)d_out;               // [B,O]

    const size_t lds_bytes = (size_t)(16 * FN_STRIDE + 16 * X_STRIDE) * sizeof(float); // 35200 B
    anfis_wmma_kernel<<<dim3(NB / 16), dim3(32), lds_bytes, stream>>>(X, Cc, Wd, Wc, Cb, Out);
}